// EDF_26731876451108
// MI455X (gfx1250) — compile-verified
//
#include <hip/hip_runtime.h>

#define NBOX 8192
#define WIMG 1920.0f
#define HIMG 1080.0f

#define AS1 __attribute__((address_space(1)))
#define AS3 __attribute__((address_space(3)))

typedef int v4i_t __attribute__((ext_vector_type(4)));

#if defined(__HIP_DEVICE_COMPILE__) && __has_builtin(__builtin_amdgcn_global_load_async_to_lds_b128)
#define ASYNC_LDS 1
#else
#define ASYNC_LDS 0
#endif

__device__ __forceinline__ void async_copy_b128(const float* g, float* l) {
#if ASYNC_LDS
  // gfx1250 async global->LDS DMA (tracked by ASYNCcnt, no VGPR round-trip)
  __builtin_amdgcn_global_load_async_to_lds_b128((AS1 v4i_t*)g, (AS3 v4i_t*)l, 0, 0);
#else
  float4 v = *(const float4*)g;
  *(float4*)l = v;
#endif
}

__device__ __forceinline__ void wait_async0() {
#if defined(__HIP_DEVICE_COMPILE__)
#if __has_builtin(__builtin_amdgcn_s_wait_asynccnt)
  __builtin_amdgcn_s_wait_asynccnt(0);
#else
  asm volatile("s_wait_asynccnt 0" ::: "memory");
#endif
#endif
}

// ---------------- Kernel 1: bitonic argsort by descending score (stable) ----
// key = (~bits(score) << 32) | idx  -> ascending u64 sort == stable desc score
__global__ void edf_sort(const float* __restrict__ scores, unsigned* __restrict__ order) {
  __shared__ unsigned long long key[NBOX];
  const int tid = threadIdx.x;  // 1024 threads
  for (int i = tid; i < NBOX; i += 1024) {
    unsigned sb = __float_as_uint(scores[i]);  // scores > 0 -> monotone bits
    key[i] = ((unsigned long long)(~sb) << 32) | (unsigned)i;
  }
  __syncthreads();
  for (int k = 2; k <= NBOX; k <<= 1) {
    for (int j = k >> 1; j > 0; j >>= 1) {
      for (int t = tid; t < NBOX / 2; t += 1024) {
        int i = ((t / j) * (2 * j)) + (t % j);
        int p = i + j;
        bool up = ((i & k) == 0);
        unsigned long long a = key[i], b = key[p];
        if ((a > b) == up) { key[i] = b; key[p] = a; }
      }
      __syncthreads();
    }
  }
  for (int i = tid; i < NBOX; i += 1024) order[i] = (unsigned)(key[i] & 0xFFFFFFFFu);
}

// ---------------- Kernel 2: gather + clip + area -----------------------------
__global__ void edf_gather(const float* __restrict__ boxes, const float* __restrict__ scores,
                           const unsigned* __restrict__ order,
                           float* __restrict__ bx1, float* __restrict__ by1,
                           float* __restrict__ bx2, float* __restrict__ by2,
                           float* __restrict__ area, float* __restrict__ s) {
  int i = blockIdx.x * blockDim.x + threadIdx.x;
  if (i >= NBOX) return;
  unsigned o = order[i];
  float x1 = fminf(fmaxf(boxes[o * 4 + 0], 0.0f), WIMG);
  float y1 = fminf(fmaxf(boxes[o * 4 + 1], 0.0f), HIMG);
  float x2 = fminf(fmaxf(boxes[o * 4 + 2], 0.0f), WIMG);
  float y2 = fminf(fmaxf(boxes[o * 4 + 3], 0.0f), HIMG);
  bx1[i] = x1; by1[i] = y1; bx2[i] = x2; by2[i] = y2;
  area[i] = (x2 - x1 + 1.0f) * (y2 - y1 + 1.0f);
  s[i] = scores[o];
}

// ---------------- Kernel 3: adjacency bitmask (IoU > 0.5) --------------------
// Block = 256 threads, 16 rows x all 8192 cols; columns staged in LDS in
// 2048-wide chunks via async global->LDS copies.
#define ROWS 16
#define CHUNK 2048
__global__ void edf_adj(const float* __restrict__ bx1, const float* __restrict__ by1,
                        const float* __restrict__ bx2, const float* __restrict__ by2,
                        const float* __restrict__ area, unsigned* __restrict__ adj) {
  __shared__ __align__(16) float cx1[CHUNK], cy1[CHUNK], cx2[CHUNK], cy2[CHUNK], car[CHUNK];
  __shared__ float rx1[ROWS], ry1[ROWS], rx2[ROWS], ry2[ROWS], rar[ROWS];
  const int tid = threadIdx.x;
  const int r0 = blockIdx.x * ROWS;
  if (tid < ROWS) {
    int r = r0 + tid;
    rx1[tid] = bx1[r]; ry1[tid] = by1[r]; rx2[tid] = bx2[r]; ry2[tid] = by2[r]; rar[tid] = area[r];
  }
  for (int ch = 0; ch < NBOX; ch += CHUNK) {
    int lo = tid * 8;            // 256 threads * 8 floats = 2048 per array
    int go = ch + lo;
    async_copy_b128(bx1 + go,     cx1 + lo);
    async_copy_b128(bx1 + go + 4, cx1 + lo + 4);
    async_copy_b128(by1 + go,     cy1 + lo);
    async_copy_b128(by1 + go + 4, cy1 + lo + 4);
    async_copy_b128(bx2 + go,     cx2 + lo);
    async_copy_b128(bx2 + go + 4, cx2 + lo + 4);
    async_copy_b128(by2 + go,     cy2 + lo);
    async_copy_b128(by2 + go + 4, cy2 + lo + 4);
    async_copy_b128(area + go,    car + lo);
    async_copy_b128(area + go + 4, car + lo + 4);
    wait_async0();
    __syncthreads();
    // 16 rows * 64 words = 1024 tasks, 4 per thread
    for (int t = tid; t < ROWS * (CHUNK / 32); t += 256) {
      int r = t >> 6;        // row in tile
      int w = t & 63;        // 32-col word within chunk
      float ax1 = rx1[r], ay1 = ry1[r], ax2 = rx2[r], ay2 = ry2[r], aa = rar[r];
      unsigned m = 0;
      int cb = w * 32;
      for (int k = 0; k < 32; ++k) {
        float ix1 = fmaxf(ax1, cx1[cb + k]);
        float iy1 = fmaxf(ay1, cy1[cb + k]);
        float iw = fminf(ax2, cx2[cb + k]) - ix1 + 1.0f;
        float ih = fminf(ay2, cy2[cb + k]) - iy1 + 1.0f;
        float inter = fmaxf(iw, 0.0f) * fmaxf(ih, 0.0f);
        float uni = aa + car[cb + k] - inter;
        m |= (inter > 0.5f * uni) ? (1u << k) : 0u;
      }
      adj[(size_t)(r0 + r) * (NBOX / 32) + (ch >> 5) + w] = m;
    }
    __syncthreads();
  }
}

// ---------------- Kernel 4: sequential greedy clustering (bit ops) -----------
__global__ void edf_cluster(const unsigned* __restrict__ adj, int* __restrict__ clusterOut) {
  __shared__ unsigned assigned[NBOX / 32];  // 256 words, one per thread
  __shared__ int clu[NBOX];
  const int tid = threadIdx.x;  // 256
  assigned[tid] = 0u;
  for (int i = tid; i < NBOX; i += 256) clu[i] = -1;
  __syncthreads();
  for (int i = 0; i < NBOX; ++i) {
    bool head = ((assigned[i >> 5] >> (i & 31)) & 1u) == 0u;
    __syncthreads();
    if (head) {
      unsigned row = adj[(size_t)i * (NBOX / 32) + tid];
      unsigned nb = row & ~assigned[tid];
      while (nb) { int b = __builtin_ctz(nb); clu[(tid << 5) + b] = i; nb &= nb - 1u; }
      assigned[tid] |= row;
    }
    __syncthreads();
  }
  for (int i = tid; i < NBOX; i += 256) clusterOut[i] = clu[i];
}

// ---------------- Kernel 5: deterministic per-cluster reductions -------------
__global__ void edf_segments(const int* __restrict__ cluster, const float* __restrict__ s,
                             const float* __restrict__ by2, float* __restrict__ probSum,
                             int* __restrict__ cnt, int* __restrict__ firstIdx) {
  const int c = blockIdx.x;
  const int tid = threadIdx.x;  // 256
  __shared__ float redF[256];
  __shared__ float redM[256];
  __shared__ int redI[256];
  __shared__ int isHead;
  if (tid == 0) isHead = (cluster[c] == c) ? 1 : 0;
  __syncthreads();
  if (!isHead) {
    if (tid == 0) { probSum[c] = 0.0f; cnt[c] = 0; firstIdx[c] = NBOX; }
    return;
  }
  float ps = 0.0f, my = -3.0e38f; int ct = 0;
  for (int j = tid; j < NBOX; j += 256)
    if (cluster[j] == c) { ps += s[j]; my = fmaxf(my, by2[j]); ct++; }
  redF[tid] = ps; redM[tid] = my; redI[tid] = ct;
  __syncthreads();
  for (int off = 128; off > 0; off >>= 1) {
    if (tid < off) {
      redF[tid] += redF[tid + off];
      redM[tid] = fmaxf(redM[tid], redM[tid + off]);
      redI[tid] += redI[tid + off];
    }
    __syncthreads();
  }
  float maxy = redM[0]; float prob = redF[0]; int count = redI[0];
  __syncthreads();
  int fi = NBOX;
  for (int j = tid; j < NBOX; j += 256)
    if (cluster[j] == c && by2[j] >= maxy) fi = min(fi, j);
  redI[tid] = fi;
  __syncthreads();
  for (int off = 128; off > 0; off >>= 1) {
    if (tid < off) redI[tid] = min(redI[tid], redI[tid + off]);
    __syncthreads();
  }
  if (tid == 0) { probSum[c] = prob; cnt[c] = count; firstIdx[c] = redI[0]; }
}

// ---------------- Kernel 6: finalize out rows + keep mask --------------------
__global__ void edf_final(const float* __restrict__ bx1, const float* __restrict__ by1,
                          const float* __restrict__ bx2, const float* __restrict__ by2,
                          const int* __restrict__ cluster, const float* __restrict__ probSum,
                          const int* __restrict__ cnt, const int* __restrict__ firstIdx,
                          const int* __restrict__ numModels, float* __restrict__ out) {
  int j = blockIdx.x * blockDim.x + threadIdx.x;
  if (j >= NBOX) return;
  int nm = numModels[0];
  int c = cluster[j];
  bool keep = (j == firstIdx[c]) && ((float)cnt[c] >= (float)nm / 3.0f);
  float sc = probSum[c] / (float)nm;
  float r0 = 0.f, r1 = 0.f, r2 = 0.f, r3 = 0.f, r4 = 0.f;
  if (keep) { r0 = bx1[j]; r1 = by1[j]; r2 = bx2[j]; r3 = by2[j]; r4 = sc; }
  out[j * 5 + 0] = r0;
  out[j * 5 + 1] = r1;
  out[j * 5 + 2] = r2;
  out[j * 5 + 3] = r3;
  out[j * 5 + 4] = r4;
  out[NBOX * 5 + j] = keep ? 1.0f : 0.0f;
}

extern "C" void kernel_launch(void* const* d_in, const int* in_sizes, int n_in,
                              void* d_out, int out_size, void* d_ws, size_t ws_size,
                              hipStream_t stream) {
  (void)in_sizes; (void)n_in; (void)out_size; (void)ws_size;
  const float* boxes = (const float*)d_in[0];
  const float* scores = (const float*)d_in[1];
  const int* numModels = (const int*)d_in[2];
  float* out = (float*)d_out;

  float* W = (float*)d_ws;
  unsigned* order = (unsigned*)(W + 0 * NBOX);
  float* bx1 = W + 1 * NBOX;
  float* by1 = W + 2 * NBOX;
  float* bx2 = W + 3 * NBOX;
  float* by2 = W + 4 * NBOX;
  float* area = W + 5 * NBOX;
  float* s   = W + 6 * NBOX;
  int* cluster = (int*)(W + 7 * NBOX);
  float* probSum = W + 8 * NBOX;
  int* cnt = (int*)(W + 9 * NBOX);
  int* firstIdx = (int*)(W + 10 * NBOX);
  unsigned* adj = (unsigned*)(W + 11 * NBOX);  // 8192*256 u32 = 8 MB

  edf_sort<<<1, 1024, 0, stream>>>(scores, order);
  edf_gather<<<NBOX / 256, 256, 0, stream>>>(boxes, scores, order, bx1, by1, bx2, by2, area, s);
  edf_adj<<<NBOX / ROWS, 256, 0, stream>>>(bx1, by1, bx2, by2, area, adj);
  edf_cluster<<<1, 256, 0, stream>>>(adj, cluster);
  edf_segments<<<NBOX, 256, 0, stream>>>(cluster, s, by2, probSum, cnt, firstIdx);
  edf_final<<<NBOX / 256, 256, 0, stream>>>(bx1, by1, bx2, by2, cluster, probSum, cnt, firstIdx,
                                            numModels, out);
}